// TextGenerationModel_20856361189690
// MI455X (gfx1250) — compile-verified
//
#include <hip/hip_runtime.h>
#include <hip/hip_bf16.h>
#include <math.h>
#include <stdint.h>

// Model dims (from reference): L=256, B=64, V=10000, E=512, H=1024
#define L_SEQ 256
#define B_SZ  64
#define V_SZ  10000
#define E_SZ  512
#define H_SZ  1024
#define H4    4096

typedef __attribute__((ext_vector_type(2))) float v2f;
typedef __attribute__((ext_vector_type(8))) float v8f;

// D = A(16x4 f32) * B(4x16 f32) + C(16x16 f32), full wave32 matrix op.
__device__ __forceinline__ v8f wmma4(v2f a, v2f b, v8f c) {
    // 8 args: (neg_a, A, neg_b, B, c_mod, C, reuse_a, reuse_b)
    return __builtin_amdgcn_wmma_f32_16x16x4_f32(
        false, a, false, b, (short)0, c, false, false);
}

// Async global->LDS copy of 16 bytes (GLOBAL_LOAD_ASYNC_TO_LDS_B128).
// LDS destination address = low 32 bits of the generic pointer (ISA aperture
// rule); global address in a 64-bit VGPR pair, GV mode.
__device__ __forceinline__ void async_copy_b128(void* lds_dst, const float* gsrc) {
    uint32_t lds_off = (uint32_t)(uintptr_t)lds_dst;
    asm volatile("global_load_async_to_lds_b128 %0, %1, off"
                 :: "v"(lds_off), "v"(gsrc)
                 : "memory");
}
#define ASYNC_WAIT_0() asm volatile("s_wait_asynccnt 0x0" ::: "memory")
#define ASYNC_WAIT_1() asm volatile("s_wait_asynccnt 0x1" ::: "memory")
#define ASYNC_WAIT_2() asm volatile("s_wait_asynccnt 0x2" ::: "memory")

__device__ __forceinline__ float sigmoidf_(float x) {
    return 1.0f / (1.0f + __expf(-x));
}

// ---------------------------------------------------------------------------
// Zero-init kernel for h0 / c0 state buffers.
// ---------------------------------------------------------------------------
__global__ __launch_bounds__(256) void k_zero(float* __restrict__ p, int n) {
    int i = blockIdx.x * blockDim.x + threadIdx.x;
    if (i < n) p[i] = 0.0f;
}

// ---------------------------------------------------------------------------
// Kernel 1: x_proj[m, 4H] = emb[x[m]] @ [Wgx|Wix|Wfx|Wox] + [bg|bi|bf|bo]
// M = L*B = 16384, K = E = 512, N = 4H = 4096.
// Workgroup: 8 waves, one 16-row tile, 8 adjacent 16-col tiles.
// A (gathered embedding rows) double-buffered in LDS via async copies.
// ---------------------------------------------------------------------------
__global__ __launch_bounds__(256) void k_embed_xproj(
    const int*   __restrict__ xIdx,
    const float* __restrict__ emb,
    const float* __restrict__ Wg, const float* __restrict__ Wi,
    const float* __restrict__ Wf, const float* __restrict__ Wo,
    const float* __restrict__ bg, const float* __restrict__ bi,
    const float* __restrict__ bf, const float* __restrict__ bo,
    float* __restrict__ xproj)
{
    const int KT = 64, KTS = 68;          // K chunk; 272B row pitch (16B-aligned)
    const int NCH = E_SZ / KT;            // 8 chunks
    __shared__ float Alds[2][16 * KTS];

    const int tid  = threadIdx.x;
    const int wave = tid >> 5;
    const int lane = tid & 31;
    const int lo   = lane & 15;
    const int hi   = lane >> 4;

    const int row0 = blockIdx.x * 16;                 // output row tile
    const int col0 = (blockIdx.y * 8 + wave) * 16;    // output col tile
    const int gate = col0 >> 10;                      // col0 / H
    const int hcol = col0 & (H_SZ - 1);

    const float* Wmat = (gate == 0) ? Wg : (gate == 1) ? Wi : (gate == 2) ? Wf : Wo;
    const float* bias = (gate == 0) ? bg : (gate == 1) ? bi : (gate == 2) ? bf : bo;

    // Cooperative A-load mapping: 256 threads x 16B = 16x64 chunk.
    const int lr = tid >> 4;            // 0..15 row within tile
    const int lc = (tid & 15) * 4;      // 0..60 col within chunk
    const float* arow = emb + (size_t)xIdx[row0 + lr] * E_SZ;

    // Prologue: chunk 0 in flight.
    async_copy_b128(&Alds[0][lr * KTS + lc], arow + lc);

    v8f acc = {};
    for (int c = 0; c < NCH; ++c) {
        const int buf = c & 1;
        if (c + 1 < NCH) {
            async_copy_b128(&Alds[buf ^ 1][lr * KTS + lc], arow + (c + 1) * KT + lc);
            ASYNC_WAIT_1();     // current chunk landed; next still in flight
        } else {
            ASYNC_WAIT_0();
        }
        __syncthreads();

        const int kk = c * KT;
#pragma unroll
        for (int k = 0; k < KT; k += 4) {
            v2f a, b;
            a.x = Alds[buf][lo * KTS + k + 2 * hi];
            a.y = Alds[buf][lo * KTS + k + 2 * hi + 1];
            const float* bp = Wmat + (size_t)(kk + k + 2 * hi) * H_SZ + hcol + lo;
            b.x = bp[0];
            b.y = bp[H_SZ];
            acc = wmma4(a, b, acc);
        }
        __syncthreads();        // reads done before buf is overwritten again
    }

    const float bv = bias[hcol + lo];
#pragma unroll
    for (int j = 0; j < 8; ++j) {
        int r = row0 + j + 8 * hi;
        xproj[(size_t)r * H4 + col0 + lo] = acc[j] + bv;
    }
}

// ---------------------------------------------------------------------------
// Kernel 2: one LSTM time step.
// z = x_proj[t] + h_prev @ Wh  (bias already folded into x_proj)
// Workgroup: 4 waves = 4 gates for one 16(batch) x 16(hcol) tile.
// h_prev K-chunk double-buffered in LDS via async copies, shared by 4 gates.
// ---------------------------------------------------------------------------
__global__ __launch_bounds__(128) void k_lstm_step(
    int t,
    const float* __restrict__ xproj,
    const float* __restrict__ Whg, const float* __restrict__ Whi,
    const float* __restrict__ Whf, const float* __restrict__ Who,
    const float* __restrict__ h_prev,
    const float* __restrict__ c_prev,
    float* __restrict__ c_cur,
    float* __restrict__ h_out)      // = h_arr + t*B*H
{
    const int KT = 64, KTS = 68;
    const int NCH = H_SZ / KT;      // 16 chunks
    __shared__ float Alds[2][16 * KTS];
    __shared__ float zlds[4][256];

    const int tid  = threadIdx.x;
    const int wave = tid >> 5;      // = gate index
    const int lane = tid & 31;
    const int lo   = lane & 15;
    const int hi   = lane >> 4;

    const int row0 = blockIdx.x * 16;   // batch tile
    const int col0 = blockIdx.y * 16;   // hidden-col tile
    const int gate = wave;

    const float* Wmat = (gate == 0) ? Whg : (gate == 1) ? Whi : (gate == 2) ? Whf : Who;

    // Cooperative A-load mapping: 128 threads x 32B (2x B128) = 16x64 chunk.
    const int lr = tid >> 3;            // 0..15
    const int lc = (tid & 7) * 8;       // 0..56
    const float* arow = h_prev + (size_t)(row0 + lr) * H_SZ;

    async_copy_b128(&Alds[0][lr * KTS + lc],     arow + lc);
    async_copy_b128(&Alds[0][lr * KTS + lc + 4], arow + lc + 4);

    v8f acc = {};
    for (int c = 0; c < NCH; ++c) {
        const int buf = c & 1;
        if (c + 1 < NCH) {
            const float* nsrc = arow + (c + 1) * KT + lc;
            async_copy_b128(&Alds[buf ^ 1][lr * KTS + lc],     nsrc);
            async_copy_b128(&Alds[buf ^ 1][lr * KTS + lc + 4], nsrc + 4);
            ASYNC_WAIT_2();
        } else {
            ASYNC_WAIT_0();
        }
        __syncthreads();

        const int kk = c * KT;
#pragma unroll
        for (int k = 0; k < KT; k += 4) {
            v2f a, b;
            a.x = Alds[buf][lo * KTS + k + 2 * hi];
            a.y = Alds[buf][lo * KTS + k + 2 * hi + 1];
            const float* bp = Wmat + (size_t)(kk + k + 2 * hi) * H_SZ + col0 + lo;
            b.x = bp[0];
            b.y = bp[H_SZ];
            acc = wmma4(a, b, acc);
        }
        __syncthreads();
    }

    // z tile -> LDS for cross-gate combine
#pragma unroll
    for (int j = 0; j < 8; ++j) {
        int r = j + 8 * hi;
        float z = acc[j] +
            xproj[(size_t)(t * B_SZ + row0 + r) * H4 + gate * H_SZ + col0 + lo];
        zlds[gate][r * 16 + lo] = z;
    }
    __syncthreads();

    // Elementwise LSTM cell update: 256 elements by 128 threads.
    for (int e = tid; e < 256; e += 128) {
        int r = e >> 4, cc = e & 15;
        float g = tanhf(zlds[0][e]);
        float i = sigmoidf_(zlds[1][e]);
        float f = sigmoidf_(zlds[2][e]);
        float o = sigmoidf_(zlds[3][e]);
        size_t idx = (size_t)(row0 + r) * H_SZ + col0 + cc;
        float cn = g * i + c_prev[idx] * f;
        c_cur[idx] = cn;
        h_out[idx] = tanhf(cn) * o;
    }
}

// ---------------------------------------------------------------------------
// Kernel 3: logits = h_arr @ linear + b_l  -> d_out (row stride 10000)
// M = 16384, K = 1024, N = 10000 = 625 exact 16-col tiles.
// ---------------------------------------------------------------------------
__global__ __launch_bounds__(256) void k_logits(
    const float* __restrict__ h_arr,
    const float* __restrict__ Wlin,
    const float* __restrict__ bl,
    float* __restrict__ out)
{
    const int KT = 64, KTS = 68;
    const int NCH = H_SZ / KT;      // 16 chunks
    __shared__ float Alds[2][16 * KTS];

    const int tid  = threadIdx.x;
    const int wave = tid >> 5;
    const int lane = tid & 31;
    const int lo   = lane & 15;
    const int hi   = lane >> 4;

    const int row0    = blockIdx.x * 16;
    const int colTile = blockIdx.y * 8 + wave;
    const bool active = colTile < (V_SZ / 16);   // 625 tiles
    const int col0    = colTile * 16;

    const int lr = tid >> 4;
    const int lc = (tid & 15) * 4;
    const float* arow = h_arr + (size_t)(row0 + lr) * H_SZ;

    async_copy_b128(&Alds[0][lr * KTS + lc], arow + lc);

    v8f acc = {};
    for (int c = 0; c < NCH; ++c) {
        const int buf = c & 1;
        if (c + 1 < NCH) {
            async_copy_b128(&Alds[buf ^ 1][lr * KTS + lc], arow + (c + 1) * KT + lc);
            ASYNC_WAIT_1();
        } else {
            ASYNC_WAIT_0();
        }
        __syncthreads();

        if (active) {
            const int kk = c * KT;
#pragma unroll
            for (int k = 0; k < KT; k += 4) {
                v2f a, b;
                a.x = Alds[buf][lo * KTS + k + 2 * hi];
                a.y = Alds[buf][lo * KTS + k + 2 * hi + 1];
                const float* bp = Wlin + (size_t)(kk + k + 2 * hi) * V_SZ + col0 + lo;
                b.x = bp[0];
                b.y = bp[V_SZ];
                acc = wmma4(a, b, acc);
            }
        }
        __syncthreads();
    }

    if (active) {
        const float bv = bl[col0 + lo];
#pragma unroll
        for (int j = 0; j < 8; ++j) {
            int r = row0 + j + 8 * hi;
            out[(size_t)r * V_SZ + col0 + lo] = acc[j] + bv;
        }
    }
}

// ---------------------------------------------------------------------------
// Kernel 4: in-place row-wise log_softmax over V=10000.
// One row per workgroup; entire row staged in LDS (40 KB of 320 KB/WGP)
// so the 655 MB logits tensor is read back exactly once.
// ---------------------------------------------------------------------------
__global__ __launch_bounds__(256) void k_logsoftmax(float* __restrict__ out)
{
    __shared__ float rowbuf[V_SZ];
    __shared__ float red[256];

    const int tid = threadIdx.x;
    float* rp = out + (size_t)blockIdx.x * V_SZ;

    float m = -3.402823466e38f;
    for (int i = tid; i < V_SZ; i += 256) {
        float v = rp[i];
        rowbuf[i] = v;
        m = fmaxf(m, v);
    }
    red[tid] = m;
    __syncthreads();
    for (int s = 128; s > 0; s >>= 1) {
        if (tid < s) red[tid] = fmaxf(red[tid], red[tid + s]);
        __syncthreads();
    }
    const float mv = red[0];
    __syncthreads();

    float sum = 0.0f;
    for (int i = tid; i < V_SZ; i += 256)
        sum += __expf(rowbuf[i] - mv);
    red[tid] = sum;
    __syncthreads();
    for (int s = 128; s > 0; s >>= 1) {
        if (tid < s) red[tid] += red[tid + s];
        __syncthreads();
    }
    const float lse = mv + logf(red[0]);

    for (int i = tid; i < V_SZ; i += 256)
        rp[i] = rowbuf[i] - lse;
}

// ---------------------------------------------------------------------------
// Host-side launcher. All launches on `stream`, graph-capture safe.
// ---------------------------------------------------------------------------
extern "C" void kernel_launch(void* const* d_in, const int* in_sizes, int n_in,
                              void* d_out, int out_size, void* d_ws, size_t ws_size,
                              hipStream_t stream)
{
    (void)in_sizes; (void)n_in; (void)out_size; (void)ws_size;

    const int*   xIdx = (const int*)  d_in[0];
    const float* emb  = (const float*)d_in[1];
    const float* Wgx  = (const float*)d_in[2];
    const float* Wgh  = (const float*)d_in[3];
    const float* bg   = (const float*)d_in[4];
    const float* Wix  = (const float*)d_in[5];
    const float* Wih  = (const float*)d_in[6];
    const float* bi   = (const float*)d_in[7];
    const float* Wfx  = (const float*)d_in[8];
    const float* Wfh  = (const float*)d_in[9];
    const float* bf   = (const float*)d_in[10];
    const float* Wox  = (const float*)d_in[11];
    const float* Woh  = (const float*)d_in[12];
    const float* bo   = (const float*)d_in[13];
    const float* lin  = (const float*)d_in[14];
    const float* bl   = (const float*)d_in[15];

    float* out = (float*)d_out;
    float* ws  = (float*)d_ws;

    // Workspace layout (floats):
    float* xproj = ws;                                           // L*B*4H = 67,108,864
    float* h_arr = xproj + (size_t)L_SEQ * B_SZ * H4;            // L*B*H  = 16,777,216
    float* h0    = h_arr + (size_t)L_SEQ * B_SZ * H_SZ;          // B*H
    float* c0    = h0 + (size_t)B_SZ * H_SZ;                     // B*H
    float* c1    = c0 + (size_t)B_SZ * H_SZ;                     // B*H

    // Zero initial h and c (h0 and c0 are contiguous: 2*B*H elements).
    {
        int n = 2 * B_SZ * H_SZ;
        k_zero<<<dim3((n + 255) / 256), 256, 0, stream>>>(h0, n);
    }

    // Embedding gather + fused 4-gate input projection (+bias).
    k_embed_xproj<<<dim3((L_SEQ * B_SZ) / 16, H4 / 128), 256, 0, stream>>>(
        xIdx, emb, Wgx, Wix, Wfx, Wox, bg, bi, bf, bo, xproj);

    // Sequential LSTM recurrence: h_prev comes straight from h_arr[t-1].
    for (int t = 0; t < L_SEQ; ++t) {
        const float* hp = (t == 0) ? h0 : (h_arr + (size_t)(t - 1) * B_SZ * H_SZ);
        const float* cp = (t & 1) ? c1 : c0;
        float*       cc = (t & 1) ? c0 : c1;
        k_lstm_step<<<dim3(B_SZ / 16, H_SZ / 16), 128, 0, stream>>>(
            t, xproj, Wgh, Wih, Wfh, Woh, hp, cp, cc,
            h_arr + (size_t)t * B_SZ * H_SZ);
    }

    // Output projection into d_out, then in-place log_softmax.
    k_logits<<<dim3((L_SEQ * B_SZ) / 16, (V_SZ / 16 + 7) / 8), 256, 0, stream>>>(
        h_arr, lin, bl, out);
    k_logsoftmax<<<dim3(L_SEQ * B_SZ), 256, 0, stream>>>(out);
}